// CBOWHierarchicalSoftmax_82454782148963
// MI455X (gfx1250) — compile-verified
//
#include <hip/hip_runtime.h>
#include <stdint.h>

// CBOW hierarchical-softmax loss on MI455X (gfx1250, wave32).
// Strategy: single workgroup. Async-gather rows straight into LDS
// (GLOBAL_LOAD_ASYNC_TO_LDS_B128, tracked by ASYNCcnt), compute the
// 17x10 Gram matrix S = N x C^T with V_WMMA_F32_16X16X4_F32 (full f32
// precision), then scores = sigmoid(rowsum(S)/10) and the summed log loss.

typedef __attribute__((ext_vector_type(2))) float v2f;
typedef __attribute__((ext_vector_type(8))) float v8f;

#define EMBED    512
#define WINDOW   10
#define PATH     17
#define LDSTRIDE (EMBED + 4)   // 516 floats/row: 516%64==4 -> conflict-free strided reads
#define LOSS_EPS 1e-9f

#if defined(__gfx1250__)
#define USE_ASYNC_LDS 1
#else
#define USE_ASYNC_LDS 0
#endif

// Copy 16 bytes global -> LDS without a VGPR round trip.
// VDST = LDS byte address (low 32 bits of the generic LDS pointer;
// ISA 10.2: flat LDS addresses truncate to addr[31:0]), VADDR = 64-bit
// global address, saddr = off. Tracked by ASYNCcnt.
__device__ __forceinline__ void async_copy16(void* lds_dst, const void* gsrc) {
#if USE_ASYNC_LDS
    uint32_t la = (uint32_t)(uintptr_t)lds_dst;
    uint64_t ga = (uint64_t)(uintptr_t)gsrc;
    asm volatile("global_load_async_to_lds_b128 %0, %1, off"
                 :: "v"(la), "v"(ga) : "memory");
#else
    float4 v = *(const float4*)gsrc;
    float* d = (float*)lds_dst;
    d[0] = v.x; d[1] = v.y; d[2] = v.z; d[3] = v.w;
#endif
}

__device__ __forceinline__ void async_wait_all() {
#if USE_ASYNC_LDS
    asm volatile("s_wait_asynccnt 0" ::: "memory");
#endif
}

__launch_bounds__(256)
__global__ void cbow_hs_wmma_kernel(const int*   __restrict__ context_idx,
                                    const int*   __restrict__ path_indices,
                                    const int*   __restrict__ code_bits,
                                    const float* __restrict__ ctx_emb,
                                    const float* __restrict__ node_emb,
                                    float*       __restrict__ out)
{
    // LDS staging: C (context rows, padded to 16 rows, rows 10..15 zero),
    // N (17 path rows), S (17x16 Gram tile), per-path loss terms.
    __shared__ float Clds[16 * LDSTRIDE];     // ~33.0 KB
    __shared__ float Nlds[PATH * LDSTRIDE];   // ~35.1 KB
    __shared__ float Sld[PATH][16];
    __shared__ float terms[32];

    const int tid = threadIdx.x;

    // --- Stage 1: gather rows into LDS (all 8 waves) -------------------
    // Zero context rows 10..15 (these become zero columns of B, so the
    // Gram-matrix columns w>=10 are exactly zero).
    for (int i = tid; i < 6 * EMBED; i += 256) {
        int r = 10 + i / EMBED, c = i % EMBED;
        Clds[r * LDSTRIDE + c] = 0.0f;
    }
    // Gather the 10 context rows, 16B per lane, direct to LDS.
    for (int i = tid; i < WINDOW * (EMBED / 4); i += 256) {
        int r = i / (EMBED / 4), c4 = i % (EMBED / 4);
        async_copy16(&Clds[r * LDSTRIDE + c4 * 4],
                     ctx_emb + (size_t)context_idx[r] * EMBED + c4 * 4);
    }
    // Gather the 17 path-node rows.
    for (int i = tid; i < PATH * (EMBED / 4); i += 256) {
        int r = i / (EMBED / 4), c4 = i % (EMBED / 4);
        async_copy16(&Nlds[r * LDSTRIDE + c4 * 4],
                     node_emb + (size_t)path_indices[r] * EMBED + c4 * 4);
    }
    async_wait_all();        // ASYNCcnt == 0: all LDS writes landed
    __syncthreads();

    // --- Stage 2: wave 0 does the WMMA Gram matrix + loss --------------
    // Lane-uniform branch: whole wave 0 enters, EXEC = all ones (WMMA req).
    if (tid < 32) {
        const int lane = tid;
        const int mh   = lane & 15;   // A: row M within tile / B,D: column N
        const int half = lane >> 4;   // selects K pair (A,B) / M+8 (D)

        v8f d0 = {};  // paths 0..15
        v8f d1 = {};  // path 16 (broadcast row; only row 0 of tile is used)

        const float* aRow0 = &Nlds[mh * LDSTRIDE];   // A tile 0: path row mh
        const float* aRow1 = &Nlds[16 * LDSTRIDE];   // A tile 1: path row 16 (bcast)
        const float* bRow  = &Clds[mh * LDSTRIDE];   // B column n=mh: context row

        // K loop: 128 steps of 16x16x4. V_WMMA_F32_16X16X4_F32 layouts:
        //   A (16x4): lane m=L&15, VGPR j = A[m][2*half + j]
        //   B (4x16): lane n=L&15, VGPR j = B[2*half + j][n]  (B[k][n] = C[n][k])
        //   D (16x16): VGPR i, lanes 0-15 -> D[i][n], lanes 16-31 -> D[i+8][n]
        #pragma unroll 8
        for (int k0 = 0; k0 < EMBED; k0 += 4) {
            const int kk = k0 + 2 * half;           // 8B-aligned (kk even)
            v2f a0 = *(const v2f*)(aRow0 + kk);
            v2f a1 = *(const v2f*)(aRow1 + kk);
            v2f b  = *(const v2f*)(bRow  + kk);
            d0 = __builtin_amdgcn_wmma_f32_16x16x4_f32(
                     false, a0, false, b, (short)0, d0, false, false);
            d1 = __builtin_amdgcn_wmma_f32_16x16x4_f32(
                     false, a1, false, b, (short)0, d1, false, false);
        }

        // Spill D tiles to LDS (same wave writes then reads: DS in-order).
        #pragma unroll
        for (int i = 0; i < 8; ++i)
            Sld[i + 8 * half][mh] = d0[i];
        if (half == 0)
            Sld[16][mh] = d1[0];                    // tile 1, row M=0 -> path 16

        // Per-path: z = rowsum/WINDOW (the mean), sigmoid, log loss term.
        float term = 0.0f;
        if (lane < PATH) {
            float s = 0.0f;
            #pragma unroll
            for (int w = 0; w < WINDOW; ++w) s += Sld[lane][w];
            float z     = s * (1.0f / (float)WINDOW);
            float score = 1.0f / (1.0f + expf(-z));
            term = (code_bits[lane] == 1) ? -logf(score + LOSS_EPS)
                                          : -logf(1.0f - score + LOSS_EPS);
        }
        terms[lane] = term;

        if (lane == 0) {
            float acc = 0.0f;
            #pragma unroll
            for (int i = 0; i < PATH; ++i) acc += terms[i];
            out[0] = acc;
        }
    }
}

extern "C" void kernel_launch(void* const* d_in, const int* in_sizes, int n_in,
                              void* d_out, int out_size, void* d_ws, size_t ws_size,
                              hipStream_t stream) {
    (void)in_sizes; (void)n_in; (void)out_size; (void)d_ws; (void)ws_size;
    const int*   context_idx  = (const int*)d_in[0];
    const int*   path_indices = (const int*)d_in[1];
    const int*   code_bits    = (const int*)d_in[2];
    const float* ctx_emb      = (const float*)d_in[3];
    const float* node_emb     = (const float*)d_in[4];
    float*       out          = (float*)d_out;

    hipLaunchKernelGGL(cbow_hs_wmma_kernel, dim3(1), dim3(256), 0, stream,
                       context_idx, path_indices, code_bits, ctx_emb, node_emb, out);
}